// lorentz_model_58944131170560
// MI455X (gfx1250) — compile-verified
//
#include <hip/hip_runtime.h>

// MI455X fused Lorentz-oscillator model.
// f32 WMMA 16x16x4 for the MLP (exact fp32 GEMM), LDS-staged padded weights
// (CDNA5 320KB/WGP LDS), per-wave 16-row tiles, fused spectrum evaluation
// with v_rcp_f32. Output-store bound (~157MB -> ~7us floor @ 23.3TB/s).
// v2: batch the 7 B-operand DS loads ahead of each WMMA burst so the
// in-order DScnt allows partial waits instead of dscnt==0 per WMMA.

typedef __attribute__((ext_vector_type(2))) float v2f;
typedef __attribute__((ext_vector_type(8))) float v8f;

#define BATCH_N   131072
#define NWAVES    8
#define ROWS_PER_BLOCK (NWAVES * 16)

// LDS layout (float offsets). Weight matrices stored "k-pair packed":
// pair index (k2, n) -> floats [2*(k2*N + n)] = {W[k=2k2][n], W[k=2k2+1][n]}
// so a WMMA B operand is one aligned float2 DS load.
#define W1T_OFF 0                    // 4*112 pairs   ->   896 floats
#define B1_OFF  896                  //                    112
#define W2T_OFF 1008                 // 56*112 pairs  -> 12544
#define B2_OFF  13552                //                    112
#define WHT_OFF 13664                // 56*16 pairs   ->  1792
#define HID_OFF 15456                // NWAVES*16*112 -> 14336
#define PST_OFF 29792                // NWAVES*16*12  ->  1536
#define SMEM_FLOATS 31328            // = 125312 bytes (2 blocks / 320KB WGP)

static __device__ inline v8f vzero8() {
  v8f z = {0.f, 0.f, 0.f, 0.f, 0.f, 0.f, 0.f, 0.f};
  return z;
}

__global__ __launch_bounds__(NWAVES * 32) void lorentz_fused(
    const float* __restrict__ G,  const float* __restrict__ W1, const float* __restrict__ b1,
    const float* __restrict__ W2, const float* __restrict__ b2,
    const float* __restrict__ Ww0, const float* __restrict__ Wwp, const float* __restrict__ Wg,
    float* __restrict__ out)
{
  extern __shared__ float smem[];
  const int tid = threadIdx.x;

  // ---- cooperative weight staging (zero-padded 100 -> 112) ----
  for (int i = tid; i < 4 * 112; i += blockDim.x) {        // W1^T : K=8, N=112
    const int k2 = i / 112, n = i % 112, k = 2 * k2;
    smem[W1T_OFF + 2 * i]     = (n < 100) ? W1[n * 8 + k]     : 0.f;
    smem[W1T_OFF + 2 * i + 1] = (n < 100) ? W1[n * 8 + k + 1] : 0.f;
  }
  for (int i = tid; i < 112; i += blockDim.x) {
    smem[B1_OFF + i] = (i < 100) ? b1[i] : 0.f;
    smem[B2_OFF + i] = (i < 100) ? b2[i] : 0.f;
  }
  for (int i = tid; i < 56 * 112; i += blockDim.x) {       // W2^T : K=112, N=112
    const int k2 = i / 112, n = i % 112, k = 2 * k2;
    const bool ok = (n < 100) && (k < 100);
    smem[W2T_OFF + 2 * i]     = ok ? W2[n * 100 + k]     : 0.f;
    smem[W2T_OFF + 2 * i + 1] = ok ? W2[n * 100 + k + 1] : 0.f;
  }
  for (int i = tid; i < 56 * 16; i += blockDim.x) {        // heads packed: cols 0-3 w0, 4-7 wp, 8-11 g
    const int k2 = i / 16, o = i % 16, k = 2 * k2;
    float x0 = 0.f, x1 = 0.f;
    if (k < 100 && o < 12) {
      const float* Wh = (o < 4) ? Ww0 : (o < 8) ? Wwp : Wg;
      const int oo = o & 3;
      x0 = Wh[oo * 100 + k];
      x1 = Wh[oo * 100 + k + 1];
    }
    smem[WHT_OFF + 2 * i]     = x0;
    smem[WHT_OFF + 2 * i + 1] = x1;
  }
  __syncthreads();

  const int wave = tid >> 5;
  const int lane = tid & 31;
  const int half = lane >> 4;     // K-half select for A/B operands
  const int l16  = lane & 15;     // A: row M; B/C/D: column N
  const long rowBase = (long)(blockIdx.x * NWAVES + wave) * 16;
  float* hid = &smem[HID_OFF + wave * 16 * 112];   // per-wave 16x112 activations

  // ---- layer 1: (16x8) @ (8x112), 2 K-steps x 7 N-tiles ----
  v8f acc[7];
#pragma unroll
  for (int nt = 0; nt < 7; ++nt) acc[nt] = vzero8();
#pragma unroll
  for (int k0 = 0; k0 < 8; k0 += 4) {
    const int k2 = (k0 >> 1) + half;               // this lane-half's K pair
    v2f a = *(const v2f*)(G + (rowBase + l16) * 8 + 2 * k2);
    v2f bb[7];
#pragma unroll
    for (int nt = 0; nt < 7; ++nt)                 // batch DS loads -> partial dscnt waits
      bb[nt] = *(const v2f*)(&smem[W1T_OFF + 2 * (k2 * 112 + nt * 16 + l16)]);
#pragma unroll
    for (int nt = 0; nt < 7; ++nt)
      acc[nt] = __builtin_amdgcn_wmma_f32_16x16x4_f32(
          false, a, false, bb[nt], (short)0, acc[nt], false, false);
  }
#pragma unroll
  for (int nt = 0; nt < 7; ++nt) {                 // bias + ReLU -> LDS (A-major)
    const int col = nt * 16 + l16;
    const float bias = smem[B1_OFF + col];
#pragma unroll
    for (int v = 0; v < 8; ++v) {
      const int m = v + 8 * half;                  // C/D row from VGPR index
      hid[m * 112 + col] = fmaxf(acc[nt][v] + bias, 0.f);
    }
  }

  // ---- layer 2: (16x112) @ (112x112), 28 K-steps x 7 N-tiles ----
  v8f acc2[7];
#pragma unroll
  for (int nt = 0; nt < 7; ++nt) acc2[nt] = vzero8();
  for (int k0 = 0; k0 < 112; k0 += 4) {
    const int k2 = (k0 >> 1) + half;
    v2f a = *(const v2f*)(&hid[l16 * 112 + 2 * k2]);
    v2f bb[7];
#pragma unroll
    for (int nt = 0; nt < 7; ++nt)
      bb[nt] = *(const v2f*)(&smem[W2T_OFF + 2 * (k2 * 112 + nt * 16 + l16)]);
#pragma unroll
    for (int nt = 0; nt < 7; ++nt)
      acc2[nt] = __builtin_amdgcn_wmma_f32_16x16x4_f32(
          false, a, false, bb[nt], (short)0, acc2[nt], false, false);
  }
#pragma unroll
  for (int nt = 0; nt < 7; ++nt) {                 // overwrite hid in place (per-wave DS order)
    const int col = nt * 16 + l16;
    const float bias = smem[B2_OFF + col];
#pragma unroll
    for (int v = 0; v < 8; ++v) {
      const int m = v + 8 * half;
      hid[m * 112 + col] = fmaxf(acc2[nt][v] + bias, 0.f);
    }
  }

  // ---- heads: (16x112) @ (112x16) one tile, 28 K-steps (4-step pipelined) ----
  v8f acch = vzero8();
  for (int k0 = 0; k0 < 112; k0 += 16) {
    v2f aa[4], bb[4];
#pragma unroll
    for (int j = 0; j < 4; ++j) {
      const int k2 = ((k0 + 4 * j) >> 1) + half;
      aa[j] = *(const v2f*)(&hid[l16 * 112 + 2 * k2]);
      bb[j] = *(const v2f*)(&smem[WHT_OFF + 2 * (k2 * 16 + l16)]);
    }
#pragma unroll
    for (int j = 0; j < 4; ++j)
      acch = __builtin_amdgcn_wmma_f32_16x16x4_f32(
          false, aa[j], false, bb[j], (short)0, acch, false, false);
  }
#pragma unroll
  for (int v = 0; v < 8; ++v) {                    // stage relu'd head outputs (cols 0..11)
    const int m = v + 8 * half;
    hid[m * 112 + l16] = fmaxf(acch[v], 0.f);
  }

  // ---- derived per-(row,osc) params: w0^2, wp^2*g, g^2 ----
  for (int p = lane; p < 64; p += 32) {
    const int r = p >> 2, o = p & 3;
    const float w0 = hid[r * 112 + o];
    const float wp = hid[r * 112 + o + 4];
    const float g  = hid[r * 112 + o + 8];
    float* ps = &smem[PST_OFF + (wave * 16 + r) * 12];
    ps[o]     = w0 * w0;
    ps[4 + o] = wp * wp * g;
    ps[8 + o] = g * g;
  }

  // ---- Lorentz spectrum: e2[b,s] = w * sum_o a_o / ((w0q_o - w^2)^2 + w^2 g2_o) ----
  const float DW = (5.0f - 0.5f) / 300.0f;
  for (int r = 0; r < 16; ++r) {
    const float* ps = &smem[PST_OFF + (wave * 16 + r) * 12];
    const float q0 = ps[0], q1 = ps[1], q2 = ps[2], q3 = ps[3];
    const float a0 = ps[4], a1 = ps[5], a2 = ps[6], a3 = ps[7];
    const float g0 = ps[8], g1 = ps[9], g2 = ps[10], g3 = ps[11];
    float* orow = out + (rowBase + r) * 300;
    for (int s = lane; s < 300; s += 32) {
      const float w  = 0.5f + DW * (float)s;
      const float w2 = w * w;
      float t, sum = 0.f;
      t = q0 - w2; sum += a0 * __builtin_amdgcn_rcpf(t * t + w2 * g0);
      t = q1 - w2; sum += a1 * __builtin_amdgcn_rcpf(t * t + w2 * g1);
      t = q2 - w2; sum += a2 * __builtin_amdgcn_rcpf(t * t + w2 * g2);
      t = q3 - w2; sum += a3 * __builtin_amdgcn_rcpf(t * t + w2 * g3);
      orow[s] = w * sum;
    }
  }
}

extern "C" void kernel_launch(void* const* d_in, const int* in_sizes, int n_in,
                              void* d_out, int out_size, void* d_ws, size_t ws_size,
                              hipStream_t stream) {
  (void)in_sizes; (void)n_in; (void)out_size; (void)d_ws; (void)ws_size;
  const float* G   = (const float*)d_in[0];
  const float* W1  = (const float*)d_in[1];
  const float* b1  = (const float*)d_in[2];
  const float* W2  = (const float*)d_in[3];
  const float* b2  = (const float*)d_in[4];
  const float* Ww0 = (const float*)d_in[5];
  const float* Wwp = (const float*)d_in[6];
  const float* Wg  = (const float*)d_in[7];
  float* out = (float*)d_out;

  dim3 grid(BATCH_N / ROWS_PER_BLOCK);     // 1024
  dim3 block(NWAVES * 32);                 // 256 threads = 8 wave32
  const size_t shmem = SMEM_FLOATS * sizeof(float);  // ~122 KB (CDNA5: 320KB/WG)
  lorentz_fused<<<grid, block, shmem, stream>>>(G, W1, b1, W2, b2, Ww0, Wwp, Wg, out);
}